// GraphConv_Customized_35871566856870
// MI455X (gfx1250) — compile-verified
//
#include <hip/hip_runtime.h>

typedef float v2f __attribute__((ext_vector_type(2)));
typedef float v8f __attribute__((ext_vector_type(8)));

#define FEATS 64
#define PSTRIDE 160   // dwords per K-pair row in LDS; 160 % 64 == 32 -> conflict-free

// ---------------- zero-init kernels (run every launch; graph-replay safe) ----
__global__ void zero_i32_kernel(int* __restrict__ p, int n) {
    int i = blockIdx.x * blockDim.x + threadIdx.x;
    if (i < n) p[i] = 0;
}
__global__ void zero_f32_kernel(float* __restrict__ p, int n) {
    int i = blockIdx.x * blockDim.x + threadIdx.x;
    if (i < n) p[i] = 0.0f;
}

// ---------------- degree accumulation (segment_sum of ones) ------------------
__global__ void degree_kernel(const int* __restrict__ src, const int* __restrict__ dst,
                              int* __restrict__ outdeg, int* __restrict__ indeg, int E) {
    int e = blockIdx.x * blockDim.x + threadIdx.x;
    if (e < E) {
        atomicAdd(&outdeg[src[e]], 1);
        atomicAdd(&indeg[dst[e]], 1);
    }
}

// ---------------- h = (feat * outdeg^-1/2) @ W via V_WMMA_F32_16X16X4_F32 ----
// 256 threads = 8 waves; each wave computes a 16x64 output tile -> 128 rows/block.
// W is staged in LDS K-pair interleaved: Wp[k>>1][col][k&1], so each B fragment
// (W[kb][c], W[kb+1][c]) is a single aligned ds_load_b64 into an even VGPR pair.
__global__ __launch_bounds__(256)
void gemm_wmma_kernel(const float* __restrict__ feat, const int* __restrict__ outdeg,
                      const float* __restrict__ W, float* __restrict__ h, int n) {
    __shared__ float Wp[32 * PSTRIDE];   // 20 KB

    const int tid = threadIdx.x;
    // cooperative load of 64x64 weight, scattered into K-pair-interleaved layout
    for (int idx = tid; idx < FEATS * FEATS; idx += 256) {
        const int k = idx >> 6;          // row of W (K index)
        const int c = idx & 63;          // column of W (N index)
        Wp[(k >> 1) * PSTRIDE + c * 2 + (k & 1)] = W[idx];
    }
    __syncthreads();

    const int wave = tid >> 5;
    const int lane = tid & 31;
    const int half = lane >> 4;       // 0: lanes 0-15, 1: lanes 16-31
    const int l16  = lane & 15;
    const int rowbase = blockIdx.x * 128 + wave * 16;
    const int row  = rowbase + l16;
    const int rowc = row < n ? row : (n - 1);      // clamp for loads only

    float deg = (float)outdeg[rowc];
    const float pre = rsqrtf(deg > 1.0f ? deg : 1.0f);
    const int koff = half * 2;                      // A/B: lanes 16-31 hold K+2

    const float* __restrict__ arow = feat + (size_t)rowc * FEATS;

    v8f acc[4];
    acc[0] = v8f{}; acc[1] = v8f{}; acc[2] = v8f{}; acc[3] = v8f{};

    #pragma unroll
    for (int kk = 0; kk < 16; ++kk) {
        const int kb = kk * 4 + koff;             // even; A/B need (kb, kb+1)
        v2f a;
        a.x = arow[kb]     * pre;
        a.y = arow[kb + 1] * pre;
        const float* __restrict__ brow = Wp + (kk * 2 + half) * PSTRIDE + l16 * 2;
        #pragma unroll
        for (int j = 0; j < 4; ++j) {
            const v2f b = *(const v2f*)(brow + j * 32);   // one ds_load_b64
            acc[j] = __builtin_amdgcn_wmma_f32_16x16x4_f32(
                false, a, false, b, (short)0, acc[j], false, false);
        }
    }

    // D layout: VGPR v, lanes 0-15 -> M=v; lanes 16-31 -> M=v+8; N = l16 + 16*j
    #pragma unroll
    for (int v = 0; v < 8; ++v) {
        const int r = rowbase + v + half * 8;
        if (r < n) {
            float* __restrict__ o = h + (size_t)r * FEATS + l16;
            o[0]  = acc[0][v];
            o[16] = acc[1][v];
            o[32] = acc[2][v];
            o[48] = acc[3][v];
        }
    }
}

// ---------------- edge scatter: agg[dst] += h[src] ---------------------------
// 16 threads per edge, float4 per thread, hardware f32 atomics (L2-resident).
__global__ __launch_bounds__(256)
void scatter_kernel(const float* __restrict__ h, const int* __restrict__ src,
                    const int* __restrict__ dst, float* __restrict__ agg, int E) {
    int t = blockIdx.x * blockDim.x + threadIdx.x;
    int e = t >> 4;
    if (e >= E) return;
    int f = (t & 15) << 2;
    int s = src[e];
    int d = dst[e];
    const float4 v = *(const float4*)(h + (size_t)s * FEATS + f);
    float* p = agg + (size_t)d * FEATS + f;
    __hip_atomic_fetch_add(p + 0, v.x, __ATOMIC_RELAXED, __HIP_MEMORY_SCOPE_AGENT);
    __hip_atomic_fetch_add(p + 1, v.y, __ATOMIC_RELAXED, __HIP_MEMORY_SCOPE_AGENT);
    __hip_atomic_fetch_add(p + 2, v.z, __ATOMIC_RELAXED, __HIP_MEMORY_SCOPE_AGENT);
    __hip_atomic_fetch_add(p + 3, v.w, __ATOMIC_RELAXED, __HIP_MEMORY_SCOPE_AGENT);
}

// ---------------- out = agg * indeg^-1/2 + bias (in place) -------------------
__global__ void finalize_kernel(float* __restrict__ out, const int* __restrict__ indeg,
                                const float* __restrict__ bias, int n) {
    int i = blockIdx.x * blockDim.x + threadIdx.x;
    if (i >= n * FEATS) return;
    int row = i >> 6;
    int f   = i & 63;
    float deg = (float)indeg[row];
    float post = rsqrtf(deg > 1.0f ? deg : 1.0f);
    out[i] = out[i] * post + bias[f];
}

extern "C" void kernel_launch(void* const* d_in, const int* in_sizes, int n_in,
                              void* d_out, int out_size, void* d_ws, size_t ws_size,
                              hipStream_t stream) {
    const float* feat = (const float*)d_in[0];
    const int*   src  = (const int*)d_in[1];
    const int*   dst  = (const int*)d_in[2];
    const float* W    = (const float*)d_in[3];
    const float* bias = (const float*)d_in[4];

    const int n = in_sizes[0] / FEATS;   // 100000
    const int E = in_sizes[1];           // 1200000

    char* ws = (char*)d_ws;
    int*   outdeg = (int*)ws;                               // n ints
    int*   indeg  = (int*)(ws + (size_t)n * 4);             // n ints
    float* h      = (float*)(ws + (size_t)n * 8);           // n*64 floats (16B aligned)
    float* out    = (float*)d_out;                          // also the atomic accumulator

    const int nf = n * FEATS;

    // zero degree counters (contiguous 2n ints) and the output accumulator
    zero_i32_kernel<<<(2 * n + 255) / 256, 256, 0, stream>>>(outdeg, 2 * n);
    zero_f32_kernel<<<(nf + 255) / 256, 256, 0, stream>>>(out, nf);

    // degrees
    degree_kernel<<<(E + 255) / 256, 256, 0, stream>>>(src, dst, outdeg, indeg, E);

    // h = (feat * pre) @ W   (WMMA f32)
    gemm_wmma_kernel<<<(n + 127) / 128, 256, 0, stream>>>(feat, outdeg, W, h, n);

    // agg[dst] += h[src]
    long tscatter = (long)E * 16;
    scatter_kernel<<<(unsigned)((tscatter + 255) / 256), 256, 0, stream>>>(h, src, dst, out, E);

    // out = agg * post + bias
    finalize_kernel<<<(nf + 255) / 256, 256, 0, stream>>>(out, indeg, bias, n);
}